// QuLinear_35235911696435
// MI455X (gfx1250) — compile-verified
//
#include <hip/hip_runtime.h>
#include <math.h>

#define NQ    14
#define BATCH 2048

typedef __attribute__((ext_vector_type(2))) float v2f;
typedef __attribute__((ext_vector_type(8))) float v8f;

// ---------------------------------------------------------------------------
// Compile-time derivation of the composed-CNOT GF(2) linear map and the
// per-output masks. The 28 CNOT perms are fixed by the reference:
//   perms = [cnot(c, (c+step)%14) for step in (1,2) for c in range(14)]
// state = state[:, perm] chained => psi'[i] = psi[L(i)],
//   L(v) = perm0(perm1(...perm27(v))), evaluated innermost (last perm) first.
// ---------------------------------------------------------------------------
struct QDerived {
  unsigned mz1[NQ];  // r_w : mask (over bit positions) for Pz product
  unsigned mx1[NQ];  // t_w : mask for Px product
  unsigned mz2[NQ], mx2[NQ], my2[NQ];  // Q1 factor selection (disjoint)
  unsigned kpack;    // popcount(t&r) & 3, 2 bits per w
};

constexpr QDerived qderive() {
  // Columns of L: col[j] = L(e_j)
  unsigned col[NQ] = {};
  for (int j = 0; j < NQ; ++j) {
    unsigned v = 1u << j;
    for (int idx = 27; idx >= 0; --idx) {          // apply last perm first
      int step = (idx < NQ) ? 1 : 2;
      int c = idx % NQ;
      int t = (c + step) % NQ;
      unsigned cm = 1u << (NQ - 1 - c);
      unsigned tm = 1u << (NQ - 1 - t);
      if (v & cm) v ^= tm;
    }
    col[j] = v;
  }
  // Rows of L
  unsigned rowL[NQ] = {};
  for (int i = 0; i < NQ; ++i) {
    unsigned r = 0;
    for (int j = 0; j < NQ; ++j) r |= ((col[j] >> i) & 1u) << j;
    rowL[i] = r;
  }
  // Invert over GF(2) (Gauss-Jordan); L is invertible by construction.
  unsigned inv[NQ] = {};
  for (int i = 0; i < NQ; ++i) inv[i] = 1u << i;
  for (int i = 0; i < NQ; ++i) {
    int p = i;
    while (p < NQ && !((rowL[p] >> i) & 1u)) ++p;
    if (p != i && p < NQ) {
      unsigned t0 = rowL[i]; rowL[i] = rowL[p]; rowL[p] = t0;
      unsigned t1 = inv[i];  inv[i]  = inv[p];  inv[p]  = t1;
    }
    for (int q = 0; q < NQ; ++q)
      if (q != i && ((rowL[q] >> i) & 1u)) { rowL[q] ^= rowL[i]; inv[q] ^= inv[i]; }
  }
  QDerived d = {};
  unsigned kp = 0;
  for (int w = 0; w < NQ; ++w) {
    int p = NQ - 1 - w;           // bit position of qubit w
    unsigned t = col[p];          // t_w = L(m_w)
    unsigned r = inv[p];          // row p of L^-1
    d.mz1[w] = r;
    d.mx1[w] = t;
    d.my2[w] = t & r;
    d.mx2[w] = t & ~r;
    d.mz2[w] = r & ~t;
    int pc = 0;
    for (int b = 0; b < NQ; ++b) pc += (int)((t & r) >> b) & 1;
    kp |= (unsigned)(pc & 3) << (2 * w);
  }
  d.kpack = kp;
  return d;
}

constexpr QDerived QD = qderive();

// ---------------------------------------------------------------------------
// One wave (32 lanes) handles 16 batch rows. Each lane builds 48 product
// values F[u] (u = w:Pz, 14+w:Px, 28+w:Q1, 42..47 pad=0) for its batch row,
// then the final out = F(16x48) @ C(48x14) contraction runs as 12 chained
// v_wmma_f32_16x16x4_f32 steps.
// A layout: M = lane%16, K = vgpr + 2*(lane/16)   (ISA 7.12.2, 32-bit A 16x4)
// B layout: N = lane%16, K = vgpr + 2*(lane/16)
// D layout: N = lane%16, M = vgpr + 8*(lane/16)
// ---------------------------------------------------------------------------
__global__ __launch_bounds__(32) void qsim_wmma_kernel(
    const float* __restrict__ x, const float* __restrict__ u3,
    float* __restrict__ out) {
  const int lane = threadIdx.x;
  const int row  = lane & 15;     // M index (batch row within tile), N index (w)
  const int hi   = lane >> 4;
  const int b    = blockIdx.x * 16 + row;

  // Per-bit-position Bloch factors; p=0..13 maps to qubit v = 13-p.
  // z = -xv/sqrt(1+xv^2); X = 1/(sqrt(1+xv^2)*sqrt(1+xv^4)); Y = xv^2 * X.
  float fz[NQ], fx[NQ], fy[NQ];
#pragma unroll
  for (int p = 0; p < NQ; ++p) {
    int v = NQ - 1 - p;
    float xv = x[b * NQ + v];
    float x2 = xv * xv;
    float i1 = 1.0f / sqrtf(1.0f + x2);
    float i2 = 1.0f / sqrtf(1.0f + x2 * x2);
    fz[p] = -xv * i1;
    fx[p] = i1 * i2;
    fy[p] = x2 * i1 * i2;
  }

  // 48 subset products (all constant-folded mask tests -> straight multiplies)
  float F[48];
#pragma unroll
  for (int w = 0; w < NQ; ++w) {
    float pz = 1.0f, px = 1.0f, q1 = 1.0f;
#pragma unroll
    for (int p = 0; p < NQ; ++p) {
      if ((QD.mz1[w] >> p) & 1u) pz *= fz[p];
      if ((QD.mx1[w] >> p) & 1u) px *= fx[p];
      if ((QD.my2[w] >> p) & 1u)      q1 *= fy[p];
      else if ((QD.mx2[w] >> p) & 1u) q1 *= fx[p];
      else if ((QD.mz2[w] >> p) & 1u) q1 *= fz[p];
    }
    F[w] = pz; F[14 + w] = px; F[28 + w] = q1;
  }
#pragma unroll
  for (int u = 42; u < 48; ++u) F[u] = 0.0f;

  // Coefficients for this lane's output column w (= lane%16); phi cancels out.
  const int w  = row;
  const int wc = (w < NQ) ? w : 0;                 // avoid OOB u3 read
  const float en  = (w < NQ) ? 1.0f : 0.0f;        // zero cols 14,15
  const float th  = u3[3 * wc + 0];
  const float la  = u3[3 * wc + 2];
  const float sth = sinf(th), cth = cosf(th);
  const float sla = sinf(la), cla = cosf(la);
  const unsigned km = (QD.kpack >> (2 * wc)) & 3u;
  const float c1 = cth * en;                        // with Pz
  const float c2 = -sth * cla * en;                 // with Px  (Re O01)
  const float mag = (km & 1u) ? (sth * sla) : (sth * cla);
  const float sgn = (km == 1u || km == 2u) ? 1.0f : -1.0f;
  const float c3 = sgn * mag * en;                  // with Q1  (Re i^k O01)

  v8f acc = {0.f, 0.f, 0.f, 0.f, 0.f, 0.f, 0.f, 0.f};
#pragma unroll
  for (int kk = 0; kk < 12; ++kk) {
    v2f a, bb;
    // A fragment: K = vgpr + 2*hi within this 4-wide K slab
    a.x = hi ? F[4 * kk + 2] : F[4 * kk + 0];
    a.y = hi ? F[4 * kk + 3] : F[4 * kk + 1];
    // B fragment: sparse coefficient matrix C[u][w]
    const int u0 = 4 * kk + 2 * hi;
    const int u1 = u0 + 1;
    bb.x = (u0 == w) ? c1 : (u0 == w + 14) ? c2 : (u0 == w + 28) ? c3 : 0.0f;
    bb.y = (u1 == w) ? c1 : (u1 == w + 14) ? c2 : (u1 == w + 28) ? c3 : 0.0f;
    acc = __builtin_amdgcn_wmma_f32_16x16x4_f32(
        /*neg_a=*/false, a, /*neg_b=*/false, bb,
        /*c_mod=*/(short)0, acc, /*reuse_a=*/false, /*reuse_b=*/false);
  }

  if (w < NQ) {
#pragma unroll
    for (int r = 0; r < 8; ++r) {
      const int m = r + 8 * hi;                    // D row = batch row in tile
      out[(blockIdx.x * 16 + m) * NQ + w] = acc[r];
    }
  }
}

extern "C" void kernel_launch(void* const* d_in, const int* in_sizes, int n_in,
                              void* d_out, int out_size, void* d_ws, size_t ws_size,
                              hipStream_t stream) {
  (void)in_sizes; (void)n_in; (void)d_ws; (void)ws_size; (void)out_size;
  const float* x  = (const float*)d_in[0];   // (2048, 14) f32
  const float* u3 = (const float*)d_in[1];   // (14, 3)   f32
  float* out = (float*)d_out;                // (2048, 14) f32
  qsim_wmma_kernel<<<BATCH / 16, 32, 0, stream>>>(x, u3, out);
}